// GCN_16904991277612
// MI455X (gfx1250) — compile-verified
//
#include <hip/hip_runtime.h>
#include <hip/hip_bf16.h>

typedef __attribute__((ext_vector_type(2))) float v2f;
typedef __attribute__((ext_vector_type(8))) float v8f;

static constexpr int N_NODES  = 100000;
static constexpr int N_EDGES  = 3200000;
static constexpr int IN_CH    = 200;
static constexpr int HID      = 64;
static constexpr int OUT_CH   = 2;
static constexpr int N_GRAPHS = 500;

// ---------------------------------------------------------------------------
// Degree: deg[col[e]] += |w[e]|   (atomic, 3.2M edges)
// ---------------------------------------------------------------------------
__global__ void gcn_deg_kernel(const int* __restrict__ col,
                               const float* __restrict__ ew,
                               float* __restrict__ deg) {
    int e = blockIdx.x * blockDim.x + threadIdx.x;
    if (e < N_EDGES) {
        atomicAdd(deg + col[e], fabsf(ew[e]));
    }
}

// dinv[n] = (deg[n] + 1)^-1/2  (in place on deg buffer)
__global__ void gcn_dinv_kernel(float* __restrict__ deg) {
    int n = blockIdx.x * blockDim.x + threadIdx.x;
    if (n < N_NODES) {
        deg[n] = rsqrtf(deg[n] + 1.0f);
    }
}

// ---------------------------------------------------------------------------
// f32 WMMA GEMM: C[M x 64] = A[M x K] * B[K x 64], row-major.
// Grid.x = M/16 tiles, block = (32,4): wave y owns n-tile y. Full EXEC.
// Uses V_WMMA_F32_16X16X4_F32 (full f32 precision; GEMM is not the
// bottleneck here so no need to drop to bf16/f16).
// ---------------------------------------------------------------------------
__global__ void gcn_gemm_wmma_f32_kernel(const float* __restrict__ A,
                                         const float* __restrict__ B,
                                         float* __restrict__ C, int K) {
    const int m0   = blockIdx.x << 4;
    const int n0   = threadIdx.y << 4;
    const int lane = threadIdx.x;
    const int l15  = lane & 15;
    const int half = lane >> 4;          // 0: K pair k..k+1, 1: K pair k+2..k+3

    v8f acc = {};
    const float* __restrict__ arow = A + (long)(m0 + l15) * K;

    for (int k = 0; k < K; k += 4) {
        const int kk = k + (half << 1);
        v2f a, b;
        // A 16x4 fragment: lane (M), VGPR0=K kk, VGPR1=K kk+1
        a.x = arow[kk];
        a.y = arow[kk + 1];
        // B 4x16 fragment: lane (N), VGPR0=K kk, VGPR1=K kk+1
        b.x = B[kk * HID + n0 + l15];
        b.y = B[(kk + 1) * HID + n0 + l15];
        acc = __builtin_amdgcn_wmma_f32_16x16x4_f32(
            /*neg_a=*/false, a, /*neg_b=*/false, b,
            /*c_mod=*/(short)0, acc, /*reuse_a=*/false, /*reuse_b=*/false);
    }

#pragma unroll
    for (int r = 0; r < 8; ++r) {
        const int row = m0 + r + (half << 3);
        C[(long)row * HID + n0 + l15] = acc[r];
    }
}

// ---------------------------------------------------------------------------
// Edge scatter: agg[col[e],:] += dinv[row]*|w|*dinv[col] * h[row[e],:]
// One wave per edge, lane handles 2 channels (float2 coalesced gather).
// row/col/w indices are wave-uniform -> scalarizable loads.
// ---------------------------------------------------------------------------
__global__ void gcn_scatter_kernel(const int* __restrict__ row,
                                   const int* __restrict__ col,
                                   const float* __restrict__ ew,
                                   const float* __restrict__ dinv,
                                   const float* __restrict__ h,
                                   float* __restrict__ agg) {
    long gid  = (long)blockIdx.x * blockDim.x + threadIdx.x;
    int  e    = (int)(gid >> 5);
    int  lane = (int)(gid & 31);
    if (e < N_EDGES) {
        int r = row[e];
        int c = col[e];
        float coeff = dinv[r] * fabsf(ew[e]) * dinv[c];
        const float2 hv = *(const float2*)(h + (long)r * HID + lane * 2);
        float* dst = agg + (long)c * HID + lane * 2;
        atomicAdd(dst,     coeff * hv.x);
        atomicAdd(dst + 1, coeff * hv.y);
    }
}

// out[n,c] = relu(agg[n,c] + dinv[n]^2 * h[n,c] + b[c])   (written over agg)
__global__ void gcn_selfloop_relu_kernel(float* __restrict__ agg,
                                         const float* __restrict__ h,
                                         const float* __restrict__ dinv,
                                         const float* __restrict__ b) {
    long i = (long)blockIdx.x * blockDim.x + threadIdx.x;
    if (i < (long)N_NODES * HID) {
        int n = (int)(i >> 6);
        int c = (int)(i & 63);
        float d  = dinv[n];
        float v  = agg[i] + d * d * h[i] + b[c];
        agg[i] = v > 0.0f ? v : 0.0f;
    }
}

// ---------------------------------------------------------------------------
// Global mean pool (sum + count via atomics; batch-index per node)
// ---------------------------------------------------------------------------
__global__ void gcn_pool_kernel(const float* __restrict__ h,
                                const int* __restrict__ batch,
                                float* __restrict__ pooled,
                                float* __restrict__ cnt) {
    long i = (long)blockIdx.x * blockDim.x + threadIdx.x;
    if (i < (long)N_NODES * HID) {
        int n = (int)(i >> 6);
        int c = (int)(i & 63);
        int g = batch[n];
        atomicAdd(pooled + (long)g * HID + c, h[i]);
        if (c == 0) atomicAdd(cnt + g, 1.0f);
    }
}

__global__ void gcn_pool_div_kernel(float* __restrict__ pooled,
                                    const float* __restrict__ cnt) {
    int i = blockIdx.x * blockDim.x + threadIdx.x;
    if (i < N_GRAPHS * HID) {
        float c = cnt[i >> 6];
        pooled[i] *= 1.0f / fmaxf(c, 1.0f);
    }
}

// x_fea[g,o] = sum_c pooled[g,c]*L1[c,o] + c1[o]    (500 x 200, K=64)
__global__ void gcn_lin1_kernel(const float* __restrict__ pooled,
                                const float* __restrict__ L1,
                                const float* __restrict__ c1,
                                float* __restrict__ xfea) {
    int i = blockIdx.x * blockDim.x + threadIdx.x;
    if (i < N_GRAPHS * IN_CH) {
        int g = i / IN_CH;
        int o = i - g * IN_CH;
        float acc = c1[o];
        const float* p = pooled + (long)g * HID;
#pragma unroll 8
        for (int c = 0; c < HID; ++c) acc += p[c] * L1[c * IN_CH + o];
        xfea[i] = acc;
    }
}

// out[g,o] = sum_k relu(xfea[g,k])*L2[k,o] + c2[o]   (500 x 2, K=200)
__global__ void gcn_lin2_kernel(const float* __restrict__ xfea,
                                const float* __restrict__ L2,
                                const float* __restrict__ c2,
                                float* __restrict__ out) {
    int i = blockIdx.x * blockDim.x + threadIdx.x;
    if (i < N_GRAPHS * OUT_CH) {
        int g = i >> 1;
        int o = i & 1;
        float acc = c2[o];
        const float* xr = xfea + (long)g * IN_CH;
        for (int k = 0; k < IN_CH; ++k) {
            float v = xr[k];
            v = v > 0.0f ? v : 0.0f;
            acc += v * L2[k * OUT_CH + o];
        }
        out[i] = acc;
    }
}

// ---------------------------------------------------------------------------
extern "C" void kernel_launch(void* const* d_in, const int* in_sizes, int n_in,
                              void* d_out, int out_size, void* d_ws, size_t ws_size,
                              hipStream_t stream) {
    const float* x    = (const float*)d_in[0];
    const int*   ei   = (const int*)d_in[1];
    const float* ea   = (const float*)d_in[2];
    const int*   bat  = (const int*)d_in[3];
    const float* W1   = (const float*)d_in[4];
    const float* b1   = (const float*)d_in[5];
    const float* W2   = (const float*)d_in[6];
    const float* b2   = (const float*)d_in[7];
    const float* L1   = (const float*)d_in[8];
    const float* c1   = (const float*)d_in[9];
    const float* L2   = (const float*)d_in[10];
    const float* c2   = (const float*)d_in[11];
    float* out = (float*)d_out;

    const int* row = ei;
    const int* col = ei + N_EDGES;

    // Workspace layout (floats)
    float* wsf    = (float*)d_ws;
    float* dinv   = wsf;                               // N_NODES
    float* bufA   = dinv + N_NODES;                    // N_NODES*HID  (h1 then h2)
    float* bufB   = bufA + (long)N_NODES * HID;        // N_NODES*HID  (agg1/a1 then agg2)
    float* pooled = bufB + (long)N_NODES * HID;        // N_GRAPHS*HID
    float* cnt    = pooled + N_GRAPHS * HID;           // N_GRAPHS
    float* xfea   = cnt + N_GRAPHS;                    // N_GRAPHS*IN_CH

    const long nodef = (long)N_NODES * HID;

    // --- degree / normalization ---
    hipMemsetAsync(dinv, 0, N_NODES * sizeof(float), stream);
    gcn_deg_kernel<<<(N_EDGES + 255) / 256, 256, 0, stream>>>(col, ea, dinv);
    gcn_dinv_kernel<<<(N_NODES + 255) / 256, 256, 0, stream>>>(dinv);

    // --- layer 1: h1 = x @ W1 (WMMA f32), scatter, self-loop+bias+relu ---
    gcn_gemm_wmma_f32_kernel<<<N_NODES / 16, dim3(32, 4), 0, stream>>>(x, W1, bufA, IN_CH);
    hipMemsetAsync(bufB, 0, nodef * sizeof(float), stream);
    {
        long threads = (long)N_EDGES * 32;
        gcn_scatter_kernel<<<(unsigned)((threads + 255) / 256), 256, 0, stream>>>(
            row, col, ea, dinv, bufA, bufB);
    }
    gcn_selfloop_relu_kernel<<<(unsigned)((nodef + 255) / 256), 256, 0, stream>>>(
        bufB, bufA, dinv, b1);

    // --- layer 2: h2 = a1 @ W2 (WMMA f32), scatter, self-loop+bias+relu ---
    gcn_gemm_wmma_f32_kernel<<<N_NODES / 16, dim3(32, 4), 0, stream>>>(bufB, W2, bufA, HID);
    hipMemsetAsync(bufB, 0, nodef * sizeof(float), stream);
    {
        long threads = (long)N_EDGES * 32;
        gcn_scatter_kernel<<<(unsigned)((threads + 255) / 256), 256, 0, stream>>>(
            row, col, ea, dinv, bufA, bufB);
    }
    gcn_selfloop_relu_kernel<<<(unsigned)((nodef + 255) / 256), 256, 0, stream>>>(
        bufB, bufA, dinv, b2);

    // --- global mean pool ---
    hipMemsetAsync(pooled, 0, (N_GRAPHS * HID + N_GRAPHS) * sizeof(float), stream);
    gcn_pool_kernel<<<(unsigned)((nodef + 255) / 256), 256, 0, stream>>>(bufB, bat, pooled, cnt);
    gcn_pool_div_kernel<<<(N_GRAPHS * HID + 255) / 256, 256, 0, stream>>>(pooled, cnt);

    // --- graph-level MLP head ---
    gcn_lin1_kernel<<<(N_GRAPHS * IN_CH + 255) / 256, 256, 0, stream>>>(pooled, L1, c1, xfea);
    gcn_lin2_kernel<<<(N_GRAPHS * OUT_CH + 255) / 256, 256, 0, stream>>>(xfea, L2, c2, out);
}